// GraphSageLayerEdgeFeat_39049842655620
// MI455X (gfx1250) — compile-verified
//
#include <hip/hip_runtime.h>
#include <math.h>

typedef __attribute__((ext_vector_type(2))) float v2f;
typedef __attribute__((ext_vector_type(8))) float v8f;

#define DD   128
#define MT   5              // 16-row M-tiles per block
#define ROWS (MT * 16)      // 80 rows per block; N = 50000 = 625 * 80

// ---------------------------------------------------------------------------
// Kernel 1: Ah = h@WA + bA, Bh = h@WB + bB  via V_WMMA_F32_16X16X4_F32.
// Block = 256 threads (8 wave32). Block owns 80 rows (5 M-tiles); wave w owns
// the 16-column tile [16w,16w+16). Each B fragment is reused across 5 M-tiles
// and 10 WMMAs issue per 4 scalar weight loads.
// ---------------------------------------------------------------------------
__global__ void __launch_bounds__(256)
gemm_ab_kernel(const float* __restrict__ h,
               const float* __restrict__ WA, const float* __restrict__ bA,
               const float* __restrict__ WB, const float* __restrict__ bB,
               float* __restrict__ Ah, float* __restrict__ Bh) {
    __shared__ float sh[ROWS * DD];             // 40 KB: 80x128 tile of h
    const int tid  = threadIdx.x;
    const int row0 = blockIdx.x * ROWS;

    // cooperative load: 80*128 = 10240 floats = 2560 float4; 10 per thread
    {
        const float4* s4 = (const float4*)(h + (size_t)row0 * DD);
        float4*       d4 = (float4*)sh;
#pragma unroll
        for (int i = 0; i < 10; ++i) d4[tid + 256 * i] = s4[tid + 256 * i];
    }
    __syncthreads();

    const int wave = tid >> 5;          // 0..7 -> column tile
    const int lane = tid & 31;
    const int half = lane >> 4;         // 0: K+0/K+1 ; 1: K+2/K+3
    const int l16  = lane & 15;         // M for A-frag, N for B/C/D frags
    const int ncol = wave * 16 + l16;

    v8f accA[MT] = {};
    v8f accB[MT] = {};

    for (int kk = 0; kk < DD; kk += 4) {
        const int ka = kk + 2 * half;
        v2f ba; ba.x = WA[(size_t)ka * DD + ncol];
                ba.y = WA[(size_t)(ka + 1) * DD + ncol];
        v2f bb; bb.x = WB[(size_t)ka * DD + ncol];
                bb.y = WB[(size_t)(ka + 1) * DD + ncol];
#pragma unroll
        for (int t = 0; t < MT; ++t) {
            v2f a; a.x = sh[(t * 16 + l16) * DD + ka];
                   a.y = sh[(t * 16 + l16) * DD + ka + 1];
            accA[t] = __builtin_amdgcn_wmma_f32_16x16x4_f32(false, a, false, ba,
                                                            (short)0, accA[t], false, false);
            accB[t] = __builtin_amdgcn_wmma_f32_16x16x4_f32(false, a, false, bb,
                                                            (short)0, accB[t], false, false);
        }
    }

    const float biasA = bA[ncol];
    const float biasB = bB[ncol];
#pragma unroll
    for (int t = 0; t < MT; ++t) {
#pragma unroll
        for (int i = 0; i < 8; ++i) {
            const int m = t * 16 + i + 8 * half;    // C/D layout
            Ah[(size_t)(row0 + m) * DD + ncol] = accA[t][i] + biasA;
            Bh[(size_t)(row0 + m) * DD + ncol] = accB[t][i] + biasB;
        }
    }
}

// ---------------------------------------------------------------------------
// Kernel 2: per-edge gated message + segment-max scatter.
// One wave per edge, one float4 per lane (coalesced 512B per array per edge).
// msg >= 0 after relu -> unsigned-bit atomicMax == float max (c init 0, which
// also implements the max(c,0) zero-in-degree clamp). Order-independent.
// ---------------------------------------------------------------------------
__global__ void __launch_bounds__(256)
edge_kernel(const int* __restrict__ src, const int* __restrict__ dst,
            const float* __restrict__ Ah, const float* __restrict__ Bh,
            unsigned int* __restrict__ c_bits, int E) {
    const int gid  = blockIdx.x * blockDim.x + threadIdx.x;
    const int e    = gid >> 5;
    const int lane = gid & 31;
    if (e >= E) return;

    const int s = src[e];
    const int d = dst[e];

    const float4 bs = ((const float4*)(Bh + (size_t)s * DD))[lane];
    const float4 bd = ((const float4*)(Bh + (size_t)d * DD))[lane];
    const float4 av = ((const float4*)(Ah + (size_t)s * DD))[lane];

    float4 m;
    m.x = fmaxf(av.x / (1.0f + __expf(-(bs.x + bd.x))), 0.0f);
    m.y = fmaxf(av.y / (1.0f + __expf(-(bs.y + bd.y))), 0.0f);
    m.z = fmaxf(av.z / (1.0f + __expf(-(bs.z + bd.z))), 0.0f);
    m.w = fmaxf(av.w / (1.0f + __expf(-(bs.w + bd.w))), 0.0f);

    unsigned int* p = c_bits + (size_t)d * DD + lane * 4;
    atomicMax(p + 0, __float_as_uint(m.x));
    atomicMax(p + 1, __float_as_uint(m.y));
    atomicMax(p + 2, __float_as_uint(m.z));
    atomicMax(p + 3, __float_as_uint(m.w));
}

// ---------------------------------------------------------------------------
// Kernel 3: bundle = [h | c] @ Wn + bn  (K = 256).
// Same WMMA scheme; K processed in two 128-wide phases (h rows of Wn, then c
// rows of Wn), restaging the 40 KB LDS tile while accumulators stay resident.
// ---------------------------------------------------------------------------
__global__ void __launch_bounds__(256)
gemm_bundle_kernel(const float* __restrict__ h, const float* __restrict__ c,
                   const float* __restrict__ Wn, const float* __restrict__ bn,
                   float* __restrict__ bundle) {
    __shared__ float sh[ROWS * DD];             // 40 KB, reused per phase
    const int tid  = threadIdx.x;
    const int row0 = blockIdx.x * ROWS;

    const int wave = tid >> 5;
    const int lane = tid & 31;
    const int half = lane >> 4;
    const int l16  = lane & 15;
    const int ncol = wave * 16 + l16;

    v8f acc[MT] = {};

    for (int phase = 0; phase < 2; ++phase) {
        const float* S = phase ? c : h;
        __syncthreads();                        // all readers done before restage
        {
            const float4* s4 = (const float4*)(S + (size_t)row0 * DD);
            float4*       d4 = (float4*)sh;
#pragma unroll
            for (int i = 0; i < 10; ++i) d4[tid + 256 * i] = s4[tid + 256 * i];
        }
        __syncthreads();

        const float* Wrow = Wn + (size_t)phase * DD * DD;   // rows 0..127 / 128..255
        for (int kk = 0; kk < DD; kk += 4) {
            const int ka = kk + 2 * half;
            v2f b; b.x = Wrow[(size_t)ka * DD + ncol];
                   b.y = Wrow[(size_t)(ka + 1) * DD + ncol];
#pragma unroll
            for (int t = 0; t < MT; ++t) {
                v2f a; a.x = sh[(t * 16 + l16) * DD + ka];
                       a.y = sh[(t * 16 + l16) * DD + ka + 1];
                acc[t] = __builtin_amdgcn_wmma_f32_16x16x4_f32(false, a, false, b,
                                                               (short)0, acc[t], false, false);
            }
        }
    }

    const float bias = bn[ncol];
#pragma unroll
    for (int t = 0; t < MT; ++t) {
#pragma unroll
        for (int i = 0; i < 8; ++i) {
            const int m = t * 16 + i + 8 * half;
            bundle[(size_t)(row0 + m) * DD + ncol] = acc[t][i] + bias;
        }
    }
}

// ---------------------------------------------------------------------------
// Kernel 4: L2 normalize per row + relu + residual. One wave per row.
// ---------------------------------------------------------------------------
__global__ void __launch_bounds__(256)
norm_kernel(const float* __restrict__ h, const float* __restrict__ bundle,
            float* __restrict__ out, int N) {
    const int gid  = blockIdx.x * blockDim.x + threadIdx.x;
    const int row  = gid >> 5;
    const int lane = gid & 31;
    if (row >= N) return;

    const float4 b = ((const float4*)(bundle + (size_t)row * DD))[lane];
    float ss = b.x * b.x + b.y * b.y + b.z * b.z + b.w * b.w;
#pragma unroll
    for (int o = 16; o > 0; o >>= 1) ss += __shfl_xor(ss, o, 32);

    const float inv = 1.0f / fmaxf(sqrtf(ss), 1e-12f);
    const float4 hv = ((const float4*)(h + (size_t)row * DD))[lane];

    float4 o4;
    o4.x = hv.x + fmaxf(b.x * inv, 0.0f);
    o4.y = hv.y + fmaxf(b.y * inv, 0.0f);
    o4.z = hv.z + fmaxf(b.z * inv, 0.0f);
    o4.w = hv.w + fmaxf(b.w * inv, 0.0f);
    ((float4*)(out + (size_t)row * DD))[lane] = o4;
}

// ---------------------------------------------------------------------------
extern "C" void kernel_launch(void* const* d_in, const int* in_sizes, int n_in,
                              void* d_out, int out_size, void* d_ws, size_t ws_size,
                              hipStream_t stream) {
    const float* h   = (const float*)d_in[0];
    const int*   src = (const int*)  d_in[1];
    const int*   dst = (const int*)  d_in[2];
    const float* WA  = (const float*)d_in[3];
    const float* bA  = (const float*)d_in[4];
    const float* WB  = (const float*)d_in[5];
    const float* bB  = (const float*)d_in[6];
    const float* Wn  = (const float*)d_in[7];
    const float* bn  = (const float*)d_in[8];
    float*       out = (float*)d_out;

    const int N = in_sizes[0] / DD;     // 50000 = 625 * 80
    const int E = in_sizes[1];          // 800000

    const size_t mat_bytes = (size_t)N * DD * sizeof(float);   // 25.6 MB
    char* ws = (char*)d_ws;
    float* Ah     = (float*)(ws + 0 * mat_bytes);
    float* Bh     = (float*)(ws + 1 * mat_bytes);
    float* cbuf   = (float*)(ws + 2 * mat_bytes);
    float* bundle = (float*)(ws + 3 * mat_bytes);

    // zero the segment-max accumulator (0 bits == 0.0f; also clamps at 0)
    hipMemsetAsync(cbuf, 0, mat_bytes, stream);

    // Phase 1: Ah, Bh   (80 rows per block)
    gemm_ab_kernel<<<N / ROWS, 256, 0, stream>>>(h, WA, bA, WB, bB, Ah, Bh);

    // Phase 2: edge messages + segment max (8 edges per 256-thread block)
    edge_kernel<<<(E + 7) / 8, 256, 0, stream>>>(src, dst, Ah, Bh,
                                                 (unsigned int*)cbuf, E);

    // Phase 3: bundle GEMM (K = 256, two 128-wide phases)
    gemm_bundle_kernel<<<N / ROWS, 256, 0, stream>>>(h, cbuf, Wn, bn, bundle);

    // Phase 4: normalize + relu + residual (8 rows per block)
    norm_kernel<<<(N + 7) / 8, 256, 0, stream>>>(h, bundle, out, N);
}